// TorchGroupwiseCost_36610301231217
// MI455X (gfx1250) — compile-verified
//
#include <hip/hip_runtime.h>
#include <hip/hip_bf16.h>

// Groupwise stereo cost volume as a banded Gram matrix via f32 WMMA.
// cost[n,g,h,w,d] = (1/8) sum_c L[n,g*8+c,h,w] * R[n,g*8+c,h,w-d], 0 if w<d.
// Band view: cost[w,d] = M[w, w-d] with M = (L/8)^T R per (n, group, h) row.

typedef __attribute__((ext_vector_type(2))) float v2f;
typedef __attribute__((ext_vector_type(8))) float v8f;

#define NB       4
#define N_GROUPS 40
#define CG       8        // channels per group
#define H_       80
#define W_       160
#define DMAX     48
#define RPAD     (DMAX + W_)   // 208 floats: 48-col zero pad in front of R row
#define THREADS  128
#define BAND_ELEMS (W_ * DMAX) // 7680

__global__ __launch_bounds__(THREADS)
void groupwise_cost_wmma(const float* __restrict__ left,
                         const float* __restrict__ right,
                         float* __restrict__ out)
{
    __shared__ float Ll[CG * W_];             //  5120 B : (L/8) group row, 8 x 160
    __shared__ float Rl[CG * RPAD];           //  6656 B : R group row, zero-padded
    __shared__ float band[BAND_ELEMS + 32];   // 30848 B : band + per-lane dump slots

    const int bid = blockIdx.x;            // (n*40+g)*80 + h
    const int h   = bid % H_;
    const int ng  = bid / H_;
    const int g   = ng % N_GROUPS;
    const int n   = ng / N_GROUPS;

    const size_t in_off = (((size_t)(n * 320 + g * CG)) * H_ + h) * W_;
    const float2* Lp2 = (const float2*)(left  + in_off);
    const float2* Rp2 = (const float2*)(right + in_off);
    const int ch2 = (H_ * W_) / 2;                          // 6400 float2

    float2* Ll2 = (float2*)Ll;   // row stride 80  float2
    float2* Rl2 = (float2*)Rl;   // row stride 104 float2

    // ---- stage rows for this (n,g,h): L pre-scaled by 1/8 (group mean),
    //      R with 48-col zero pad so cost(d > w) naturally evaluates to 0.
    for (int idx = threadIdx.x; idx < CG * (W_ / 2); idx += THREADS) {
        const int c  = idx / (W_ / 2);
        const int w2 = idx - c * (W_ / 2);
        float2 lv = Lp2[c * ch2 + w2];
        float2 rv = Rp2[c * ch2 + w2];
        lv.x *= 0.125f; lv.y *= 0.125f;
        Ll2[c * (W_ / 2) + w2]          = lv;
        Rl2[c * (RPAD / 2) + 24 + w2]   = rv;
    }
    for (int idx = threadIdx.x; idx < CG * (DMAX / 2); idx += THREADS) {
        const int c  = idx / (DMAX / 2);
        const int w2 = idx - c * (DMAX / 2);
        Rl2[c * (RPAD / 2) + w2] = make_float2(0.0f, 0.0f);
    }
    __syncthreads();

    // ---- band Gram: wave j covers disparity super-tile d ~ [16j-15, 16j+15]
    const int lane = threadIdx.x & 31;
    const int j    = threadIdx.x >> 5;   // 0..3
    const int hi   = lane >> 4;          // lane half (K / M-row split)
    const int m    = lane & 15;          // A row M, B col N, D col N
    const int kA   = 2 * hi;             // K for vgpr.x; vgpr.y = K+1

    // D lane layout: M = v + 8*hi, N = m  ->  w = w0+M, d = 16j + M - m.
    // Per-lane: d(v) = d0 + v, elem addr(v) = base + v*49 (Δw=1 & Δd=1).
    const int d0    = 16 * j + 8 * hi - m;
    const int dumpA = BAND_ELEMS + lane;            // never read back

    for (int i = 0; i < 10; ++i) {
        const int w0 = 16 * i;                 // output w-tile base
        const int rb = w0 - 16 * j + DMAX;     // padded R base, 0..192

        // A: 16x4 f32 (M=w, K=c), lane m row M=m, K = 2*hi + {0,1} across vgprs
        // B: 4x16 f32 (K=c, N=w'), lane m col N=m, same K striping
        v2f a0, a1, b0, b1;
        a0.x = Ll[(kA + 0) * W_ + w0 + m];
        a0.y = Ll[(kA + 1) * W_ + w0 + m];
        a1.x = Ll[(kA + 4) * W_ + w0 + m];
        a1.y = Ll[(kA + 5) * W_ + w0 + m];
        b0.x = Rl[(kA + 0) * RPAD + rb + m];
        b0.y = Rl[(kA + 1) * RPAD + rb + m];
        b1.x = Rl[(kA + 4) * RPAD + rb + m];
        b1.y = Rl[(kA + 5) * RPAD + rb + m];

        // K=8 reduction = two chained K=4 f32 WMMAs
        v8f acc = {};
        acc = __builtin_amdgcn_wmma_f32_16x16x4_f32(
                  false, a0, false, b0, (short)0, acc, false, false);
        acc = __builtin_amdgcn_wmma_f32_16x16x4_f32(
                  false, a1, false, b1, (short)0, acc, false, false);

        // Unconditional scatter: invalid d redirected to per-lane dump slot
        // (v_cndmask instead of exec-mask branches). Every valid (w,d) cell
        // is produced exactly once across (j, i, v).
        const int base = (w0 + 8 * hi) * DMAX + d0;
        #pragma unroll
        for (int v = 0; v < 8; ++v) {
            const int d = d0 + v;
            const int a = ((unsigned)d < (unsigned)DMAX) ? (base + v * 49) : dumpA;
            band[a] = acc[v];
        }
    }
    __syncthreads();

    // ---- coalesced vectorized writeback: band layout == output layout
    float4*       ob = (float4*)(out + (size_t)bid * BAND_ELEMS);
    const float4* bb = (const float4*)band;
    for (int idx = threadIdx.x; idx < BAND_ELEMS / 4; idx += THREADS) {
        ob[idx] = bb[idx];
    }
}

extern "C" void kernel_launch(void* const* d_in, const int* in_sizes, int n_in,
                              void* d_out, int out_size, void* d_ws, size_t ws_size,
                              hipStream_t stream) {
    (void)in_sizes; (void)n_in; (void)out_size; (void)d_ws; (void)ws_size;
    const float* left  = (const float*)d_in[0];
    const float* right = (const float*)d_in[1];
    float* out = (float*)d_out;

    dim3 grid(NB * N_GROUPS * H_);   // 12800 blocks, one per (n,g,h)
    dim3 block(THREADS);             // 4 waves: one per disparity super-tile
    groupwise_cost_wmma<<<grid, block, 0, stream>>>(left, right, out);
}